// Submodule_75196287418545
// MI455X (gfx1250) — compile-verified
//
#include <hip/hip_runtime.h>

typedef float v2f __attribute__((ext_vector_type(2)));
typedef float v8f __attribute__((ext_vector_type(8)));

#define NB      64      // batch
#define SMAX    2048
#define PAGE_T  128
#define NH      32
#define NKVH    8
#define HDIM    128
#define GSZ     (NH / NKVH)   // 4
#define NWAVE   4
#define QPAD    132           // LDS row stride for Q (bank-conflict pad)
#define ATTN_SCALE 0.08838834764831845f  // 1/sqrt(128)

__global__ __launch_bounds__(128) void paged_gqa_decode_f32wmma(
    const int*   __restrict__ req_to_token,
    const int*   __restrict__ req_pool_indices,
    const int*   __restrict__ seq_lens,
    const float* __restrict__ query,
    const float* __restrict__ k_cache,
    const float* __restrict__ v_cache,
    float*       __restrict__ out)
{
    __shared__ float qs[GSZ * QPAD];
    __shared__ float red_m[NWAVE][GSZ];
    __shared__ float red_l[NWAVE][GSZ];
    __shared__ float red_o[NWAVE][GSZ][HDIM];

    const int bkv  = blockIdx.x;
    const int b    = bkv / NKVH;
    const int kvh  = bkv % NKVH;
    const int tid  = threadIdx.x;
    const int lane = tid & 31;
    const int wave = tid >> 5;

    const int r  = req_pool_indices[b];
    const int Ls = seq_lens[b];

    // Stage the 4 query rows for this (b, kvh) into LDS.
    for (int i = tid; i < GSZ * HDIM; i += blockDim.x) {
        const int g = i / HDIM, d = i % HDIM;
        qs[g * QPAD + d] = query[((size_t)(b * NH + kvh * GSZ + g)) * HDIM + d];
    }
    __syncthreads();

    const int hi = lane >> 4;          // half-wave select (K-chunk split)
    const int ln = lane & 15;          // row/col within 16
    const int gq = (ln < GSZ) ? ln : (GSZ - 1);  // clamp cols 4-15 -> dup of q3 (ignored)

    const size_t kv_row = (size_t)NKVH * HDIM;   // floats between consecutive tokens

    v8f oacc[8];
    const v8f vzero = {0.f, 0.f, 0.f, 0.f, 0.f, 0.f, 0.f, 0.f};
#pragma unroll
    for (int i = 0; i < 8; ++i) oacc[i] = vzero;
    float m_run = -1e30f;
    float l_run = 0.0f;

    // Flash-decoding: each wave owns 16-token tiles strided by 4 tiles.
    for (int t = wave; t * 16 < Ls; t += NWAVE) {
        const int s0   = t * 16;
        const int page = req_to_token[r * SMAX + (s0 / PAGE_T) * PAGE_T] / PAGE_T;
        const int poff = s0 % PAGE_T;
        const size_t base = ((size_t)(page * PAGE_T + poff) * NKVH + kvh) * HDIM;
        const float* krow  = k_cache + base + (size_t)ln * kv_row; // token s0+ln row
        const float* vbase = v_cache + base;                       // token s0 row

        // ---- scores tile: D[16 tok x 16 col], cols 0-3 = GQA heads ----
        v8f dacc = vzero;
#pragma unroll
        for (int i = 0; i < 32; ++i) {
            const int d0 = 4 * i;
            v2f a = *reinterpret_cast<const v2f*>(krow + d0 + 2 * hi);          // K[tok][d0+2hi..+1]
            v2f qb = *reinterpret_cast<const v2f*>(&qs[gq * QPAD + d0 + 2 * hi]); // Q[g][d0+2hi..+1]
            dacc = __builtin_amdgcn_wmma_f32_16x16x4_f32(
                false, a, false, qb, (short)0, dacc, false, false);
        }

        // ---- masked scale + online softmax (per query = per lane) ----
        float p[8];
        float m_half = -1e30f;
#pragma unroll
        for (int v = 0; v < 8; ++v) {
            const int tok = s0 + v + 8 * hi;
            const float s = (tok < Ls) ? dacc[v] * ATTN_SCALE : -1e30f;
            p[v] = s;
            m_half = fmaxf(m_half, s);
        }
        const float m_tile = fmaxf(m_half, __shfl(m_half, lane ^ 16, 32));
        const float m_new  = fmaxf(m_run, m_tile);
        const float alpha  = __expf(m_run - m_new);
        float s_half = 0.0f;
#pragma unroll
        for (int v = 0; v < 8; ++v) {
            p[v] = __expf(p[v] - m_new);
            s_half += p[v];
        }
        l_run = l_run * alpha + s_half + __shfl(s_half, lane ^ 16, 32);
        m_run = m_new;

        // ---- build P B-fragments (4 token-chunks of 4) via lane permutes ----
        // P[tok][q] lives at p[tok&7], source lane = q + 16*(tok>>3)
        v2f bf[4];
#pragma unroll
        for (int c = 0; c < 4; ++c) {
            const int k0 = 4 * c;
            const float v0lo = __shfl(p[(k0 + 0) & 7], ln | (((k0 + 0) >> 3) << 4), 32);
            const float v0hi = __shfl(p[(k0 + 2) & 7], ln | (((k0 + 2) >> 3) << 4), 32);
            const float v1lo = __shfl(p[(k0 + 1) & 7], ln | (((k0 + 1) >> 3) << 4), 32);
            const float v1hi = __shfl(p[(k0 + 3) & 7], ln | (((k0 + 3) >> 3) << 4), 32);
            v2f bb;
            bb.x = (lane < 16) ? v0lo : v0hi;
            bb.y = (lane < 16) ? v1lo : v1hi;
            bf[c] = bb;
        }

        // ---- O[d][q] += V^T x P : 8 head-dim tiles x 4 token-chunks ----
#pragma unroll
        for (int dt = 0; dt < 8; ++dt) {
            v8f o = oacc[dt];
#pragma unroll
            for (int j = 0; j < 8; ++j) o[j] *= alpha;   // rescale running accum
#pragma unroll
            for (int c = 0; c < 4; ++c) {
                const int k0 = 4 * c;
                const float* vr = vbase + (size_t)(k0 + 2 * hi) * kv_row + dt * 16 + ln;
                v2f a;
                a.x = vr[0];        // V[tok k0+2hi  ][dt*16+ln]
                a.y = vr[kv_row];   // V[tok k0+2hi+1][dt*16+ln]
                o = __builtin_amdgcn_wmma_f32_16x16x4_f32(
                    false, a, false, bf[c], (short)0, o, false, false);
            }
            oacc[dt] = o;
        }
    }

    // ---- per-wave partials -> LDS ----
    if (ln < GSZ) {
        if (hi == 0) {
            red_m[wave][ln] = m_run;
            red_l[wave][ln] = l_run;
        }
#pragma unroll
        for (int dt = 0; dt < 8; ++dt)
#pragma unroll
            for (int v = 0; v < 8; ++v)
                red_o[wave][ln][dt * 16 + v + 8 * hi] = oacc[dt][v];
    }
    __syncthreads();

    // ---- cross-wave flash combine + store ----
    for (int idx = tid; idx < GSZ * HDIM; idx += blockDim.x) {
        const int g = idx / HDIM, d = idx % HDIM;
        float mt = -1e30f;
#pragma unroll
        for (int w = 0; w < NWAVE; ++w) mt = fmaxf(mt, red_m[w][g]);
        float lt = 0.0f, acc = 0.0f;
#pragma unroll
        for (int w = 0; w < NWAVE; ++w) {
            const float sc = __expf(red_m[w][g] - mt);
            lt  += sc * red_l[w][g];
            acc += sc * red_o[w][g][d];
        }
        out[((size_t)(b * NH + kvh * GSZ + g)) * HDIM + d] = acc / lt;
    }
}

extern "C" void kernel_launch(void* const* d_in, const int* in_sizes, int n_in,
                              void* d_out, int out_size, void* d_ws, size_t ws_size,
                              hipStream_t stream) {
    const int*   req_to_token     = (const int*)d_in[0];
    const int*   req_pool_indices = (const int*)d_in[1];
    const int*   seq_lens         = (const int*)d_in[2];
    const float* query            = (const float*)d_in[3];
    const float* k_cache          = (const float*)d_in[4];
    const float* v_cache          = (const float*)d_in[5];
    float*       out              = (float*)d_out;

    dim3 grid(NB * NKVH);   // one block per (batch, kv_head)
    dim3 block(128);        // 4 wave32s
    paged_gqa_decode_f32wmma<<<grid, block, 0, stream>>>(
        req_to_token, req_pool_indices, seq_lens, query, k_cache, v_cache, out);
}